// MultiPower_77902116815084
// MI455X (gfx1250) — compile-verified
//
#include <hip/hip_runtime.h>

#define N 8192

typedef __attribute__((ext_vector_type(2))) float v2f;
typedef __attribute__((ext_vector_type(8))) float v8f;

// ---------------------------------------------------------------------------
// Stage 1: prefix scan + per-row/per-column precompute.  One block, 256 thr,
// 32 elements per thread.
// ---------------------------------------------------------------------------
__global__ __launch_bounds__(256) void mp_stage1(
    const float* __restrict__ step, const float* __restrict__ eta,
    const float* __restrict__ pA, const float* __restrict__ pAlpha,
    const float* __restrict__ pGamma, const float* __restrict__ pL0,
    float* __restrict__ ps, float* __restrict__ dlr,
    float* __restrict__ sc, float* __restrict__ ct,
    float* __restrict__ out)
{
    __shared__ float tot[256];
    const int tid  = threadIdx.x;
    const int base = tid * 32;

    const float A     = *pA;
    const float alpha = *pAlpha;
    const float gamma = *pGamma;
    const float L0    = *pL0;

    // local inclusive scan of t[n] = 0.5 * eta[n] * (step[n+1]-step[n-1]) (clamped)
    float loc[32];
    float run = 0.0f;
    #pragma unroll
    for (int k = 0; k < 32; ++k) {
        int n  = base + k;
        int np = (n + 1 < N) ? n + 1 : N - 1;
        int nm = (n - 1 > 0) ? n - 1 : 0;
        float sd = step[np] - step[nm];
        run += 0.5f * eta[n] * sd;
        loc[k] = run;
    }
    tot[tid] = run;
    __syncthreads();

    // Hillis-Steele inclusive scan over the 256 per-thread totals
    for (int off = 1; off < 256; off <<= 1) {
        float v   = tot[tid];
        float add = (tid >= off) ? tot[tid - off] : 0.0f;
        __syncthreads();
        tot[tid] = v + add;
        __syncthreads();
    }

    const float base_add = 0.5f * eta[0] * step[0];
    const float offset   = ((tid > 0) ? tot[tid - 1] : 0.0f) + base_add;

    #pragma unroll
    for (int k = 0; k < 32; ++k) {
        int n = base + k;
        float p = loc[k] + offset;               // partial_sum[n] > 0
        ps[n] = p;
        float ctv = A * __builtin_amdgcn_exp2f(-alpha * __builtin_amdgcn_logf(p)) + L0;
        ct[n] = ctv;
        if (n == 0) out[0] = ctv;                // loss_drop[0] == 0
        if (n <= N - 2) {
            dlr[n] = eta[n] - eta[n + 1];        // diff_lr
            sc[n]  = __builtin_amdgcn_exp2f(-gamma * __builtin_amdgcn_logf(eta[n + 1]));
        } else {
            dlr[n] = 0.0f;                       // zero pad (feeds WMMA A harmlessly)
            sc[n]  = 1.0f;
        }
    }
}

// ---------------------------------------------------------------------------
// Stage 2: triangular reduction, one wave per 16-column tile.
//   loss_drop[j] = B * sum_{i<=j} dlr[i] * (1 - (1 + C*sc[j]*(ps[j+1]-ps[i]))^-beta)
// WMMA f32 16x16x4: A broadcast = dlr[i0..i0+3] in every row, B = 4x16 power
// tile. All D rows identical; lane n (0..15) acc[0] = column jb+n result.
// ---------------------------------------------------------------------------
__global__ __launch_bounds__(32) void mp_stage2(
    const float* __restrict__ ps, const float* __restrict__ dlr,
    const float* __restrict__ sc, const float* __restrict__ ct,
    const float* __restrict__ pB, const float* __restrict__ pC,
    const float* __restrict__ pBeta,
    float* __restrict__ out)
{
    const int lane = threadIdx.x;            // 0..31, single wave per block
    const int nloc = lane & 15;              // column within tile
    const int kb   = (lane >> 4) << 1;       // K base: 0 (lanes 0-15) or 2 (16-31)
    const int jb   = blockIdx.x * 16;
    const int j    = jb + nloc;
    const bool validj = (j <= N - 2);
    const int jc   = validj ? j : (N - 2);   // clamp for safe loads

    const float Bp    = *pB;
    const float Cp    = *pC;
    const float beta  = *pBeta;
    const float pscol = ps[jc + 1];
    const float csc   = Cp * sc[jc];

    v8f acc = {0.f, 0.f, 0.f, 0.f, 0.f, 0.f, 0.f, 0.f};

    const int imax = jb + 16;                // covers i up to jb+15 (>= max j in tile)
    for (int i0 = 0; i0 < imax; i0 += 4) {
        const int ia = i0 + kb;              // this lane's two K rows
        const int ib = ia + 1;               // ib <= jb+15 <= 8191 (in bounds)

        float d0 = dlr[ia];
        float d1 = dlr[ib];
        float r0 = ps[ia];
        float r1 = ps[ib];

        v2f a;                               // A broadcast across all M rows
        a.x = d0; a.y = d1;

        float x0 = fmaf(csc, pscol - r0, 1.0f);
        float x1 = fmaf(csc, pscol - r1, 1.0f);
        // power = 1 - x^(-beta); x>0 wherever unmasked
        float p0 = 1.0f - __builtin_amdgcn_exp2f(-beta * __builtin_amdgcn_logf(x0));
        float p1 = 1.0f - __builtin_amdgcn_exp2f(-beta * __builtin_amdgcn_logf(x1));

        v2f b;                               // mask below diagonal BEFORE WMMA (NaN-safe)
        b.x = (validj && ia <= j) ? p0 : 0.0f;
        b.y = (validj && ib <= j) ? p1 : 0.0f;

        acc = __builtin_amdgcn_wmma_f32_16x16x4_f32(
            /*neg_a=*/false, a, /*neg_b=*/false, b,
            /*c_mod=*/(short)0, acc, /*reuse_a=*/false, /*reuse_b=*/false);
    }

    // All D rows identical; lanes 0-15: acc[0] = (M=0, N=lane) result.
    if (lane < 16 && validj) {
        out[j + 1] = ct[j + 1] - Bp * acc[0];
    }
}

// ---------------------------------------------------------------------------
extern "C" void kernel_launch(void* const* d_in, const int* in_sizes, int n_in,
                              void* d_out, int out_size, void* d_ws, size_t ws_size,
                              hipStream_t stream) {
    const float* step  = (const float*)d_in[0];
    const float* eta   = (const float*)d_in[1];
    const float* A     = (const float*)d_in[2];
    const float* B     = (const float*)d_in[3];
    const float* C     = (const float*)d_in[4];
    const float* alpha = (const float*)d_in[5];
    const float* beta  = (const float*)d_in[6];
    const float* gamma = (const float*)d_in[7];
    const float* L0    = (const float*)d_in[8];
    float* out = (float*)d_out;

    float* ps  = (float*)d_ws;     // [N]
    float* dlr = ps  + N;          // [N] (zero-padded at N-1)
    float* sc  = dlr + N;          // [N]
    float* ct  = sc  + N;          // [N]

    mp_stage1<<<1, 256, 0, stream>>>(step, eta, A, alpha, gamma, L0,
                                     ps, dlr, sc, ct, out);
    mp_stage2<<<512, 32, 0, stream>>>(ps, dlr, sc, ct, B, C, beta, out);
}